// DetectModelGRUAdapt_56126632624644
// MI455X (gfx1250) — compile-verified
//
#include <hip/hip_runtime.h>
#include <hip/hip_bf16.h>
#include <math.h>

// ---------------- problem constants ----------------
#define TN_   65536          // B*N nodes
#define TE_   262144         // B*E_PER edges
#define B_    128
#define N_    512
#define D_    16
#define NREL_ 114
#define CAP_  (TE_ + NREL_*16)   // relation-padded sorted-edge capacity
#define TILES_CAP_ (CAP_/16)
#define NEG_SLOPE_ 0.2f

typedef __attribute__((ext_vector_type(2))) float v2f;
typedef __attribute__((ext_vector_type(8))) float v8f;

// order-preserving float<->uint encoding for atomicMax on signed floats
__device__ __forceinline__ unsigned fenc(float f) {
  unsigned u = __float_as_uint(f);
  return (u & 0x80000000u) ? ~u : (u | 0x80000000u);
}
__device__ __forceinline__ float fdec(unsigned e) {
  unsigned u = (e & 0x80000000u) ? (e ^ 0x80000000u) : ~e;
  return __uint_as_float(u);
}

// ---------------- setup kernels ----------------
__global__ void k_init_all(int* cnt, int* cur, int* sortedE, int* tileRel,
                           int* nmask, int* emask) {
  int i = blockIdx.x * blockDim.x + threadIdx.x;
  if (i < CAP_)       sortedE[i] = -1;
  if (i < TE_)        emask[i] = 1;
  if (i < TN_)        nmask[i] = 1;
  if (i < TILES_CAP_) tileRel[i] = -1;
  if (i < NREL_)      { cnt[i] = 0; cur[i] = 0; }
}

__global__ void k_hist(const int* __restrict__ etype, int* cnt) {
  int e = blockIdx.x * blockDim.x + threadIdx.x;
  if (e < TE_) atomicAdd(&cnt[etype[e]], 1);
}

__global__ void k_scan_tiles(const int* __restrict__ cnt, int* offs, int* tileRel) {
  if (blockIdx.x == 0 && threadIdx.x == 0) {
    int o = 0;
    for (int r = 0; r < NREL_; ++r) {
      offs[r] = o;
      int padded = (cnt[r] + 15) & ~15;
      int t0 = o >> 4, nt = padded >> 4;
      for (int t = 0; t < nt; ++t) tileRel[t0 + t] = r;
      o += padded;
    }
    offs[NREL_] = o;
  }
}

__global__ void k_scatter(const int* __restrict__ etype, const int* __restrict__ offs,
                          int* cur, int* sortedE) {
  int e = blockIdx.x * blockDim.x + threadIdx.x;
  if (e >= TE_) return;
  int r = etype[e];
  int pos = offs[r] + atomicAdd(&cur[r], 1);
  sortedE[pos] = e;
}

__global__ void k_embed(const float* __restrict__ emb, const int* __restrict__ attr,
                        float* __restrict__ x) {
  int i = blockIdx.x * blockDim.x + threadIdx.x;   // over TN*16
  if (i >= TN_ * D_) return;
  int node = i >> 4, d = i & 15;
  x[i] = emb[attr[node] * D_ + d];
}

__global__ void k_layer_init(float* outacc, unsigned* maxenc, float* den) {
  int i = blockIdx.x * blockDim.x + threadIdx.x;   // over TN*16
  if (i >= TN_ * D_) return;
  outacc[i] = 0.f;
  if (i < TN_) { maxenc[i] = fenc(-__builtin_inff()); den[i] = 0.f; }
}

// ---------------- WMMA relational transform ----------------
// One wave (32 lanes) per 16-edge tile; all edges in a tile share relation r.
// h_s[16x16] = A_src[16x16] @ W_r[16x16], h_t likewise, via 4 chained
// V_WMMA_F32_16X16X4_F32 per output matrix (K split 16 = 4x4).
//
// D-row m depends only on A-row m, so padded slots (edge id < 0) are handled
// by CLAMPING indices to 0 (unconditional vector loads -> full EXEC, no
// saveexec branches) and steering their stores to a dump buffer.
__global__ __launch_bounds__(32)
void k_transform_wmma(const float* __restrict__ x, const float* __restrict__ Wl,
                      const int* __restrict__ sortedE, const int* __restrict__ tileRel,
                      const int* __restrict__ srcI, const int* __restrict__ dstI,
                      float* __restrict__ hs, float* __restrict__ ht,
                      float* __restrict__ dump) {
  int t = blockIdx.x;
  int r = tileRel[t];
  if (r < 0) return;
  int lane = threadIdx.x;
  int m    = lane & 15;          // A row / D column index
  int hi   = lane >> 4;          // half-wave select
  int koff = hi * 2;             // K sub-offset within each 4-chunk
  int n    = m;                  // B/D column for this lane

  // per-lane edge metadata for A-row m (lanes l and l+16 duplicate; fine)
  int e_m = sortedE[t * 16 + m];
  int ec  = (e_m < 0) ? 0 : e_m;               // clamped edge id
  int sm  = srcI[ec];
  int dm  = dstI[ec];

  const float* W = Wl + (size_t)r * (D_ * D_);
  v2f b[4];
#pragma unroll
  for (int c = 0; c < 4; ++c) {
    int k0 = 4 * c + koff;
    b[c].x = W[(k0 + 0) * D_ + n];
    b[c].y = W[(k0 + 1) * D_ + n];
  }

  const v2f* xs = (const v2f*)(x + (size_t)sm * D_);
  const v2f* xt = (const v2f*)(x + (size_t)dm * D_);
  v8f accS = {}; v8f accT = {};
#pragma unroll
  for (int c = 0; c < 4; ++c) {
    v2f a = xs[(4 * c + koff) >> 1];           // global_load_b64, unconditional
    accS = __builtin_amdgcn_wmma_f32_16x16x4_f32(false, a, false, b[c],
                                                 (short)0, accS, false, false);
  }
#pragma unroll
  for (int c = 0; c < 4; ++c) {
    v2f a = xt[(4 * c + koff) >> 1];
    accT = __builtin_amdgcn_wmma_f32_16x16x4_f32(false, a, false, b[c],
                                                 (short)0, accT, false, false);
  }

  // store: lane holds D column n of rows (hi*8 .. hi*8+7)
#pragma unroll
  for (int v = 0; v < 8; ++v) {
    int row = v + hi * 8;
    int er  = __shfl(e_m, row, 32);            // lane 'row' holds edge id of row
    // unconditional store; invalid rows go to dump (cndmask address select)
    float* ps = (er >= 0) ? (hs + (size_t)er * D_ + n) : (dump + lane);
    float* pt = (er >= 0) ? (ht + (size_t)er * D_ + n) : (dump + 32 + lane);
    *ps = accS[v];
    *pt = accT[v];
  }
}

// ---------------- attention kernels ----------------
__global__ void k_logits(const float* __restrict__ hs, const float* __restrict__ ht,
                         const float* __restrict__ q, const float* __restrict__ kk,
                         const int* __restrict__ dstI, const int* __restrict__ emask,
                         float* __restrict__ logit, unsigned* maxenc) {
  int e = blockIdx.x * blockDim.x + threadIdx.x;
  if (e >= TE_) return;
  float acc = 0.f;
#pragma unroll
  for (int d = 0; d < D_; ++d)
    acc += ht[(size_t)e * D_ + d] * q[d] + hs[(size_t)e * D_ + d] * kk[d];
  float lg = (acc > 0.f) ? acc : NEG_SLOPE_ * acc;
  logit[e] = lg;
  if (emask[e]) atomicMax(&maxenc[dstI[e]], fenc(lg));
}

__global__ void k_exp(const float* __restrict__ logit, const unsigned* __restrict__ maxenc,
                      const int* __restrict__ dstI, const int* __restrict__ emask,
                      float* __restrict__ ex, float* __restrict__ den) {
  int e = blockIdx.x * blockDim.x + threadIdx.x;
  if (e >= TE_) return;
  float v = 0.f;
  if (emask[e]) {
    float mv = fdec(maxenc[dstI[e]]);
    float mm = (mv >= -3.0e38f) ? mv : 0.f;   // isfinite substitute
    v = __expf(logit[e] - mm);
  }
  ex[e] = v;
  if (v != 0.f) atomicAdd(&den[dstI[e]], v);
}

__global__ void k_aggregate(const float* __restrict__ hs, const float* __restrict__ ex,
                            const float* __restrict__ den, const int* __restrict__ dstI,
                            const int* __restrict__ emask, float* __restrict__ outacc) {
  int e = blockIdx.x * blockDim.x + threadIdx.x;
  if (e >= TE_ || !emask[e]) return;
  float exv = ex[e];
  if (exv == 0.f) return;
  int dn = dstI[e];
  float dd = den[dn];
  float alpha = exv / ((dd > 0.f) ? dd : 1.f);
#pragma unroll
  for (int d = 0; d < D_; ++d)
    atomicAdd(&outacc[(size_t)dn * D_ + d], alpha * hs[(size_t)e * D_ + d]);
}

__global__ void k_finalize(const float* __restrict__ outacc, const float* __restrict__ bias,
                           const int* __restrict__ nmask, float* __restrict__ x) {
  int i = blockIdx.x * blockDim.x + threadIdx.x;   // over TN*16
  if (i >= TN_ * D_) return;
  int node = i >> 4, d = i & 15;
  float v = fmaxf(outacc[i] + bias[d], 0.f);
  x[i] = nmask[node] ? v : 0.f;
}

// ---------------- pooling / head ----------------
__global__ void k_pool(const float* __restrict__ x, const int* __restrict__ nmask,
                       float* __restrict__ g, int layer) {
  int b = blockIdx.x, d = threadIdx.x;
  if (d >= D_) return;
  float sum = 0.f, mx = -3.0e38f;
  int cnt = 0;
  for (int i = 0; i < N_; ++i) {
    int gi = b * N_ + i;
    float v = x[(size_t)gi * D_ + d];
    sum += v;                               // masked rows are zero
    if (nmask[gi]) { mx = fmaxf(mx, v); ++cnt; }
  }
  g[b * 96 + layer * 32 + d]       = sum / (float)cnt;
  g[b * 96 + layer * 32 + 16 + d]  = mx;
}

__global__ __launch_bounds__(512)
void k_topk(float* __restrict__ x, int* __restrict__ nmask,
            const float* __restrict__ pw, int keep_k) {
  int b = blockIdx.x, i = threadIdx.x;
  int gi = b * N_ + i;
  __shared__ float s[N_];
  float nrm2 = 0.f;
#pragma unroll
  for (int d = 0; d < D_; ++d) nrm2 += pw[d] * pw[d];
  float inv = rsqrtf(nrm2);
  float dotv = 0.f;
#pragma unroll
  for (int d = 0; d < D_; ++d) dotv += x[(size_t)gi * D_ + d] * pw[d];
  float sc = tanhf(dotv * inv);
  s[i] = nmask[gi] ? sc : -3.0e38f;
  __syncthreads();
  float mine = s[i];
  int rank = 0;
  for (int j = 0; j < N_; ++j) {
    float o = s[j];
    rank += (o > mine) || (o == mine && j < i);   // matches top_k index tiebreak
  }
  int keep = (rank < keep_k) && nmask[gi];
#pragma unroll
  for (int d = 0; d < D_; ++d)
    x[(size_t)gi * D_ + d] = keep ? x[(size_t)gi * D_ + d] * sc : 0.f;
  nmask[gi] = keep;
}

__global__ void k_edge_mask(const int* __restrict__ srcI, const int* __restrict__ dstI,
                            const int* __restrict__ nmask, int* __restrict__ emask) {
  int e = blockIdx.x * blockDim.x + threadIdx.x;
  if (e >= TE_) return;
  emask[e] = emask[e] && nmask[srcI[e]] && nmask[dstI[e]];
}

__global__ __launch_bounds__(128)
void k_mlp(const float* __restrict__ g, const float* __restrict__ w1, const float* __restrict__ b1,
           const float* __restrict__ w2, const float* __restrict__ b2,
           const float* __restrict__ w3, const float* __restrict__ b3,
           float* __restrict__ out) {
  int b = threadIdx.x;
  if (b >= B_) return;
  float h1[16];
  for (int j = 0; j < 16; ++j) {
    float a = b1[j];
    for (int i = 0; i < 96; ++i) a += g[b * 96 + i] * w1[i * 16 + j];
    h1[j] = fmaxf(a, 0.f);
  }
  float h2[4];
  for (int j = 0; j < 4; ++j) {
    float a = b2[j];
    for (int i = 0; i < 16; ++i) a += h1[i] * w2[i * 4 + j];
    h2[j] = fmaxf(a, 0.f);
  }
  float z = b3[0];
  for (int i = 0; i < 4; ++i) z += h2[i] * w3[i];
  out[b] = 1.f / (1.f + __expf(-z));
}

// ---------------- host launcher ----------------
extern "C" void kernel_launch(void* const* d_in, const int* in_sizes, int n_in,
                              void* d_out, int out_size, void* d_ws, size_t ws_size,
                              hipStream_t stream) {
  const float* emb    = (const float*)d_in[0];
  const float* gnn_w  = (const float*)d_in[1];   // [3][114][16][16]
  const float* gnn_q  = (const float*)d_in[2];   // [3][16][1]
  const float* gnn_k  = (const float*)d_in[3];
  const float* gnn_b  = (const float*)d_in[4];   // [3][16]
  const float* pool_w = (const float*)d_in[5];   // [2][16]
  const float* w1 = (const float*)d_in[6];
  const float* b1 = (const float*)d_in[7];
  const float* w2 = (const float*)d_in[8];
  const float* b2 = (const float*)d_in[9];
  const float* w3 = (const float*)d_in[10];
  const float* b3 = (const float*)d_in[11];
  const int* x_attr     = (const int*)d_in[12];
  const int* edge_index = (const int*)d_in[13];  // [2][TE]
  const int* edge_type  = (const int*)d_in[14];
  const int* srcI = edge_index;
  const int* dstI = edge_index + TE_;
  float* out = (float*)d_out;

  char* w = (char*)d_ws;
  size_t o = 0;
  auto carve = [&](size_t bytes) -> void* {
    void* p = w + o;
    o = (o + bytes + 255) & ~(size_t)255;
    return p;
  };
  float*    x       = (float*)carve((size_t)TN_ * D_ * 4);
  float*    outacc  = (float*)carve((size_t)TN_ * D_ * 4);
  float*    hs      = (float*)carve((size_t)TE_ * D_ * 4);
  float*    ht      = (float*)carve((size_t)TE_ * D_ * 4);
  float*    logit   = (float*)carve((size_t)TE_ * 4);
  float*    ex      = (float*)carve((size_t)TE_ * 4);
  unsigned* maxenc  = (unsigned*)carve((size_t)TN_ * 4);
  float*    den     = (float*)carve((size_t)TN_ * 4);
  int*      nmask   = (int*)carve((size_t)TN_ * 4);
  int*      emask   = (int*)carve((size_t)TE_ * 4);
  int*      sortedE = (int*)carve((size_t)CAP_ * 4);
  int*      tileRel = (int*)carve((size_t)TILES_CAP_ * 4);
  int*      cnt     = (int*)carve(NREL_ * 4);
  int*      cur     = (int*)carve(NREL_ * 4);
  int*      offs    = (int*)carve((NREL_ + 1) * 4);
  float*    g       = (float*)carve((size_t)B_ * 96 * 4);
  float*    dump    = (float*)carve(64 * 4);

  const int T256 = 256;
  int gCap = (CAP_ + T256 - 1) / T256;
  int gTE  = (TE_ + T256 - 1) / T256;
  int gTND = (TN_ * D_ + T256 - 1) / T256;

  // one-time (per call) edge bucketing by relation
  k_init_all<<<gCap, T256, 0, stream>>>(cnt, cur, sortedE, tileRel, nmask, emask);
  k_hist<<<gTE, T256, 0, stream>>>(edge_type, cnt);
  k_scan_tiles<<<1, 1, 0, stream>>>(cnt, offs, tileRel);
  k_scatter<<<gTE, T256, 0, stream>>>(edge_type, offs, cur, sortedE);
  k_embed<<<gTND, T256, 0, stream>>>(emb, x_attr, x);

  const int keep_ks[2] = {410, 328};   // ceil(0.8*512), ceil(0.8*410)
  for (int l = 0; l < 3; ++l) {
    const float* Wl = gnn_w + (size_t)l * NREL_ * D_ * D_;
    const float* ql = gnn_q + l * D_;
    const float* kl = gnn_k + l * D_;
    const float* bl = gnn_b + l * D_;

    k_layer_init<<<gTND, T256, 0, stream>>>(outacc, maxenc, den);
    k_transform_wmma<<<TILES_CAP_, 32, 0, stream>>>(x, Wl, sortedE, tileRel,
                                                    srcI, dstI, hs, ht, dump);
    k_logits<<<gTE, T256, 0, stream>>>(hs, ht, ql, kl, dstI, emask, logit, maxenc);
    k_exp<<<gTE, T256, 0, stream>>>(logit, maxenc, dstI, emask, ex, den);
    k_aggregate<<<gTE, T256, 0, stream>>>(hs, ex, den, dstI, emask, outacc);
    k_finalize<<<gTND, T256, 0, stream>>>(outacc, bl, nmask, x);
    k_pool<<<B_, 16, 0, stream>>>(x, nmask, g, l);
    if (l < 2) {
      k_topk<<<B_, N_, 0, stream>>>(x, nmask, pool_w + l * D_, keep_ks[l]);
      k_edge_mask<<<gTE, T256, 0, stream>>>(srcI, dstI, nmask, emask);
    }
  }
  k_mlp<<<1, 128, 0, stream>>>(g, w1, b1, w2, b2, w3, b3, out);
}